// UncertaintyAwareLoss_37185826848899
// MI455X (gfx1250) — compile-verified
//
#include <hip/hip_runtime.h>

typedef float v2f __attribute__((ext_vector_type(2)));
typedef float v8f __attribute__((ext_vector_type(8)));

#define UAL_EPS 1e-8f

// Compute one 16x16 f32 Gram tile: A-tile rows (from LDS) x B-tile rows (from
// global, used as columns) over K=128, via 32 chained V_WMMA_F32_16X16X4_F32.
// Fragment layout (32-bit A 16x4): lanes 0-15 hold K=0,1 ; lanes 16-31 hold
// K=2,3 -> per lane a float2 at (idx16*128 + 4k + 2*half). B mirrors it.
__device__ __forceinline__ v8f gram_tile(const float* __restrict__ zcol,
                                         const float* __restrict__ Atile,
                                         int idx16, int half) {
    const v2f* __restrict__ arow = (const v2f*)(Atile + idx16 * 128 + 2 * half);
    const v2f* __restrict__ brow = (const v2f*)(zcol + idx16 * 128 + 2 * half);
    v8f acc = {};
#pragma unroll
    for (int k = 0; k < 32; ++k) {
        v2f a = arow[2 * k];  // floats at 4k+2*half, 4k+2*half+1
        v2f b = brow[2 * k];
        acc = __builtin_amdgcn_wmma_f32_16x16x4_f32(
            /*neg_a=*/false, a, /*neg_b=*/false, b,
            /*c_mod=*/(short)0, acc, /*reuse_a=*/false, /*reuse_b=*/false);
    }
    return acc;
}

__global__ __launch_bounds__(256)
void ual_main_kernel(const float* __restrict__ z,      // [4096,128]
                     const int* __restrict__ labels,   // [256]
                     const float* __restrict__ unc,    // [4096]
                     const int* __restrict__ epoch_p,  // [1]
                     float* __restrict__ partial)      // [256] (ws)
{
    __shared__ float Atile[16 * 128];   // anchor row tile, row-major
    __shared__ float Sneg[16];
    __shared__ float Qacc[16];
    __shared__ float RowLoss[16];
    __shared__ float SumNeg[16];
    __shared__ float Lam[16];
    __shared__ int   Labs[256];

    const int r     = blockIdx.x;       // row tile id == batch element id
    const int tid   = threadIdx.x;
    const int wave  = tid >> 5;
    const int lane  = tid & 31;
    const int half  = lane >> 4;
    const int idx16 = lane & 15;

    // Stage labels + anchor tile + per-row state.
    Labs[tid] = labels[tid];
    {
        const float* src = z + (size_t)r * 2048;
#pragma unroll
        for (int i = 0; i < 8; ++i)
            Atile[tid + 256 * i] = src[tid + 256 * i];
    }
    if (tid < 16) {
        Sneg[tid] = 0.0f;
        Qacc[tid] = 0.0f;
        RowLoss[tid] = 0.0f;
        const float ep = (float)epoch_p[0];
        const float u  = unc[r * 16 + tid];
        Lam[tid] = 1.0f - u * (1.0f - ep / 100.0f);
    }
    __syncthreads();

    const int lr = Labs[r];

    // ---- Sweep 1: Sneg_i = sum(e*neg), Q_i = sum((e*neg)^2) over all cols ----
    float snegL[8], qL[8];
#pragma unroll
    for (int v = 0; v < 8; ++v) { snegL[v] = 0.0f; qL[v] = 0.0f; }

    for (int t = 0; t < 32; ++t) {
        const int  c    = wave + 8 * t;   // column tile handled by this wave
        const int  lc   = Labs[c];
        const bool own  = (c == r);
        const bool same = (lc == lr);
        v8f acc = gram_tile(z + (size_t)c * 2048, Atile, idx16, half);
#pragma unroll
        for (int v = 0; v < 8; ++v) {
            const int  m     = v + 8 * half;          // row within tile
            const bool eyeel = own && (idx16 == m);   // global i == j
            // neg_mask = !(same_class & ~eye)
            const float negm = (!same || eyeel) ? 1.0f : 0.0f;
            const float e  = expf(acc[v] / 0.1f);
            const float en = e * negm;                // keeps inf*0 -> NaN like ref
            snegL[v] += en;
            qL[v]    += en * en;
        }
    }
    // Reduce across the 16 lanes of each half (rows m=v and m=v+8).
#pragma unroll
    for (int v = 0; v < 8; ++v) {
        float s = snegL[v], q = qL[v];
        for (int msk = 1; msk <= 8; msk <<= 1) {
            s += __shfl_xor(s, msk, 32);
            q += __shfl_xor(q, msk, 32);
        }
        if (idx16 == 0) {
            atomicAdd(&Sneg[v + 8 * half], s);
            atomicAdd(&Qacc[v + 8 * half], q);
        }
    }
    __syncthreads();
    if (tid < 16) SumNeg[tid] = Qacc[tid] / (Sneg[tid] + UAL_EPS);
    __syncthreads();

    // ---- Sweep 2: row_loss_i = sum_pos -log(num/(num+SumNeg_i+eps)+eps) ----
    float lossL[8];
#pragma unroll
    for (int v = 0; v < 8; ++v) lossL[v] = 0.0f;

    for (int t = 0; t < 32; ++t) {
        const int  c    = wave + 8 * t;
        const int  lc   = Labs[c];
        const bool own  = (c == r);
        const bool same = (lc == lr);
        v8f acc = gram_tile(z + (size_t)c * 2048, Atile, idx16, half);
#pragma unroll
        for (int v = 0; v < 8; ++v) {
            const int  m     = v + 8 * half;
            const bool eyeel = own && (idx16 == m);
            const float posm = (same && !eyeel) ? 1.0f : 0.0f;
            const float e    = expf(acc[v] / 0.1f);
            const float num  = e * (Lam[m] * posm);   // exp_sim * (lam * pos_mask)
            const float den  = num + SumNeg[m] + UAL_EPS;
            lossL[v] += -logf(num / den + UAL_EPS) * posm;
        }
    }
#pragma unroll
    for (int v = 0; v < 8; ++v) {
        float s = lossL[v];
        for (int msk = 1; msk <= 8; msk <<= 1) s += __shfl_xor(s, msk, 32);
        if (idx16 == 0) atomicAdd(&RowLoss[v + 8 * half], s);
    }
    __syncthreads();

    if (tid == 0) {
        int cnt = 0;
        for (int c = 0; c < 256; ++c) cnt += (Labs[c] == lr) ? 1 : 0;
        const float npos = 16.0f * (float)cnt - 1.0f;   // same for all 16 rows
        float s = 0.0f;
        for (int m = 0; m < 16; ++m)
            s += RowLoss[m] / (npos + UAL_EPS);          // npos >= 15 -> valid
        partial[r] = s;
    }
}

__global__ __launch_bounds__(256)
void ual_finalize_kernel(const float* __restrict__ partial,
                         float* __restrict__ out) {
    __shared__ float buf[256];
    const int t = threadIdx.x;
    buf[t] = partial[t];
    __syncthreads();
    for (int s = 128; s > 0; s >>= 1) {
        if (t < s) buf[t] += buf[t + s];
        __syncthreads();
    }
    if (t == 0) out[0] = buf[0] / 4096.0f;   // n_valid == M (every row valid)
}

extern "C" void kernel_launch(void* const* d_in, const int* in_sizes, int n_in,
                              void* d_out, int out_size, void* d_ws, size_t ws_size,
                              hipStream_t stream) {
    (void)in_sizes; (void)n_in; (void)out_size; (void)ws_size;
    const float* z      = (const float*)d_in[0];
    const int*   labels = (const int*)d_in[1];
    const float* unc    = (const float*)d_in[2];
    const int*   epoch  = (const int*)d_in[3];
    float* partial = (float*)d_ws;   // 256 floats
    float* out     = (float*)d_out;

    ual_main_kernel<<<256, 256, 0, stream>>>(z, labels, unc, epoch, partial);
    ual_finalize_kernel<<<1, 256, 0, stream>>>(partial, out);
}